// SpatialDiffusion_76132590289373
// MI455X (gfx1250) — compile-verified
//
#include <hip/hip_runtime.h>

#define NNODE 50000
#define NEDGE 800000
#define DIM   96

typedef __attribute__((ext_vector_type(2))) float v2f;
typedef __attribute__((ext_vector_type(8))) float v8f;

// ---------------------------------------------------------------------------
// Kernel 0: zero the workspace (agg[N*96] + deg[N]) every launch.
// ---------------------------------------------------------------------------
__global__ void zero_ws_kernel(float* __restrict__ ws, int n) {
  int i = blockIdx.x * blockDim.x + threadIdx.x;
  if (i < n) ws[i] = 0.0f;
}

// ---------------------------------------------------------------------------
// Kernel 1: edge scatter. One wave32 per edge. 96 feats = 3 per lane.
// agg[dst] += x[src]; deg[dst] += 1. All of agg/x fit in the 192MB L2, so
// these atomics resolve at L2 throughput.
// ---------------------------------------------------------------------------
__global__ void __launch_bounds__(256)
scatter_kernel(const float* __restrict__ x,
               const int*   __restrict__ src,
               const int*   __restrict__ dst,
               float*       __restrict__ agg,
               float*       __restrict__ deg) {
  int wave = (blockIdx.x * blockDim.x + threadIdx.x) >> 5;
  int lane = threadIdx.x & 31;
  if (wave >= NEDGE) return;  // wave-uniform guard

  int s = src[wave];
  int d = dst[wave];
  const float* xs = x   + (long)s * DIM;
  float*       ad = agg + (long)d * DIM;

#pragma unroll
  for (int c = 0; c < 3; ++c) {
    int f = lane + 32 * c;
    atomicAdd(&ad[f], xs[f]);
  }
  if (lane == 0) atomicAdd(&deg[d], 1.0f);
}

// ---------------------------------------------------------------------------
// Kernel 2: fused (agg/deg) @ Wl^T + x @ Wr^T + bl, ReLU.
// One wave per 16x16 output tile; K swept in steps of 4 with
// V_WMMA_F32_16X16X4_F32. 50000 % 16 == 0 and 96 % 16 == 0, so no bounds
// guards and EXEC is all-1s for every WMMA (ISA requirement).
//
// f32 WMMA fragment layout (ISA 7.12.2):
//   A 16x4 : lane = {half,m}: m=lane&15 is M; v0=A[m][k+2*half], v1=A[m][k+2*half+1]
//   B 4x16 : lane = {half,n}: n=lane&15 is N; v0=B[k+2*half][n],  v1=B[k+2*half+1][n]
//   C/D    : VGPR v, lanes 0-15: (M=v,   N=lane); lanes 16-31: (M=v+8, N=lane-16)
// B[k][n] = W[n][k] (W is [DOUT,DIN] row-major), so both A and B fragments are
// two consecutive floats -> single b64 loads.
// ---------------------------------------------------------------------------
__global__ void __launch_bounds__(256)
sage_gemm_kernel(const float* __restrict__ agg,
                 const float* __restrict__ deg,
                 const float* __restrict__ x,
                 const float* __restrict__ Wl,
                 const float* __restrict__ bl,
                 const float* __restrict__ Wr,
                 float*       __restrict__ out) {
  const int NT = (NNODE / 16) * (DIM / 16);  // 3125 * 6 = 18750 tiles
  int wave = (blockIdx.x * blockDim.x + threadIdx.x) >> 5;
  if (wave >= NT) return;  // wave-uniform: EXEC remains all-1s in live waves

  int lane     = threadIdx.x & 31;
  int row_tile = wave / (DIM / 16);
  int col_tile = wave % (DIM / 16);
  int m        = lane & 15;   // M for A-frag, N for B-frag & C/D column
  int half     = lane >> 4;   // selects K pair {0,1} vs {2,3}

  int row = row_tile * 16 + m;   // node row this lane loads for A
  int cn  = col_tile * 16 + m;   // output column this lane loads for B

  float rdeg = 1.0f / fmaxf(deg[row], 1.0f);

  const float* aggr = agg + (long)row * DIM;
  const float* xr   = x   + (long)row * DIM;
  const float* wlc  = Wl  + (long)cn  * DIM;
  const float* wrc  = Wr  + (long)cn  * DIM;

  // Bias is per output column N: preload into the accumulator.
  float bias = bl[cn];
  v8f acc;
#pragma unroll
  for (int i = 0; i < 8; ++i) acc[i] = bias;

#pragma unroll
  for (int kb = 0; kb < DIM / 4; ++kb) {
    int kk = kb * 4 + half * 2;

    // Neighbor-mean path: A = agg/deg, B = Wl^T
    v2f a, b;
    a.x = aggr[kk] * rdeg;
    a.y = aggr[kk + 1] * rdeg;
    b.x = wlc[kk];
    b.y = wlc[kk + 1];
    acc = __builtin_amdgcn_wmma_f32_16x16x4_f32(
        /*neg_a=*/false, a, /*neg_b=*/false, b,
        /*c_mod=*/(short)0, acc, /*reuse_a=*/false, /*reuse_b=*/false);

    // Root path: A = x, B = Wr^T
    v2f a2, b2;
    a2.x = xr[kk];
    a2.y = xr[kk + 1];
    b2.x = wrc[kk];
    b2.y = wrc[kk + 1];
    acc = __builtin_amdgcn_wmma_f32_16x16x4_f32(
        false, a2, false, b2, (short)0, acc, false, false);
  }

  // ReLU + store. C/D layout: VGPR v -> M = v + 8*half, N = m.
  long out_base = (long)(row_tile * 16) * DIM + col_tile * 16 + m;
#pragma unroll
  for (int v = 0; v < 8; ++v) {
    int M = v + 8 * half;
    out[out_base + (long)M * DIM] = fmaxf(acc[v], 0.0f);
  }
}

// ---------------------------------------------------------------------------
extern "C" void kernel_launch(void* const* d_in, const int* in_sizes, int n_in,
                              void* d_out, int out_size, void* d_ws, size_t ws_size,
                              hipStream_t stream) {
  const float* x  = (const float*)d_in[0];
  const int*   ei = (const int*)d_in[1];   // [2, E] row-major: src then dst
  const float* Wl = (const float*)d_in[2];
  const float* bl = (const float*)d_in[3];
  const float* Wr = (const float*)d_in[4];
  float* out = (float*)d_out;

  float* agg = (float*)d_ws;                       // N * 96 floats
  float* deg = agg + (size_t)NNODE * DIM;          // N floats

  int nzero = NNODE * DIM + NNODE;
  zero_ws_kernel<<<(nzero + 255) / 256, 256, 0, stream>>>(agg, nzero);

  // one wave32 per edge: E*32 threads
  long sthreads = (long)NEDGE * 32;
  scatter_kernel<<<(int)((sthreads + 255) / 256), 256, 0, stream>>>(
      x, ei, ei + NEDGE, agg, deg);

  // one wave32 per 16x16 tile: 18750 waves
  long gthreads = (long)(NNODE / 16) * (DIM / 16) * 32;
  sage_gemm_kernel<<<(int)((gthreads + 255) / 256), 256, 0, stream>>>(
      agg, deg, x, Wl, bl, Wr, out);
}